// signature_model_35502199668985
// MI455X (gfx1250) — compile-verified
//
#include <hip/hip_runtime.h>

typedef __attribute__((ext_vector_type(2))) float v2f;
typedef __attribute__((ext_vector_type(8))) float v8f;

#define CC 20
#define LL 1024
#define CH 64
#define CHP 65           // stride 65 == 1 (mod 64 banks): conflict-free
#define NCHUNK 16        // 1024 padded increments (1023 real + 1 zero)
#define SIGDIM (CC + CC*CC + CC*CC*CC)   // 8420
#define NOUT 10

// LDS layout (floats). SIG/O10 alias the chunk arrays (dead after last K loop).
#define OFF_D   0
#define OFF_R   (32*CHP)        // 2080
#define OFF_A   (2*32*CHP)      // 4160
#define OFF_B   (3*32*CHP)      // 6240
#define CHUNK_FLOATS (4*32*CHP) // 8320
#define OFF_SIG 0               // alias
#define OFF_OUT SIGDIM          // 8420
#define SMEM_FLOATS (SIGDIM + 16)  // 8436 floats = 33744 B (< 64KB cap)

__global__ __launch_bounds__(256)
void sig3_kernel(const float* __restrict__ X,
                 const float* __restrict__ W,
                 const float* __restrict__ bias,
                 float* __restrict__ out) {
  extern __shared__ float smem[];
  float* DB  = smem + OFF_D;    // d_t  [32][CHP], rows>=20 stay zero
  float* RB  = smem + OFF_R;    // R_t
  float* AB  = smem + OFF_A;    // A_t
  float* BB  = smem + OFF_B;    // B_t
  float* SIG = smem + OFF_SIG;  // aliased signature [8420]
  float* O10 = smem + OFF_OUT;  // 10 output partials (+pad)

  const int tid = threadIdx.x;
  const int b   = blockIdx.x;
  const float* Xb = X + (size_t)b * LL * CC;

  // zero chunk arrays once (pad rows 20..31 must be zero for every chunk)
  for (int i = tid; i < CHUNK_FLOATS; i += 256) smem[i] = 0.0f;

  // ---- fill-role mapping: 64 timesteps x 4 channel-groups of 5 ----
  const int tloc = tid >> 2;     // 0..63 timestep within chunk
  const int grp  = tid & 3;
  const int c0   = grp * 5;
  float xf[5], xl[5];
  #pragma unroll
  for (int cc = 0; cc < 5; ++cc) {
    xf[cc] = Xb[c0 + cc];
    xl[cc] = Xb[(LL - 1) * CC + c0 + cc];
  }

  // ---- wave / lane / tile setup ----
  const int w    = tid >> 5;
  const int lid  = tid & 31;
  const int r    = lid & 15;     // tile row/col within 16
  const int half = lid >> 4;     // K half (lanes 16-31 hold K=2,3)

  // slots 0..5: always S3 tiles (tau = w+8s <= 47 < 50). slot 6: mixed.
  int iOff[7], jOff[7], mbase[7], ntA[7];
  #pragma unroll
  for (int s = 0; s < 6; ++s) {
    int tau = w + 8 * s;
    int mt = tau >> 1, nt = tau & 1;
    int m = mt * 16 + r;
    int i = m / 20, j = m - 20 * i;
    iOff[s] = i * CHP; jOff[s] = j * CHP;
    mbase[s] = mt * 16; ntA[s] = nt;
  }
  int tau6 = w + 48;                 // 48..55
  if (tau6 >= 54) tau6 -= 4;         // dup 50/51: benign identical writes
  const int slot6_is3 = (tau6 < 50); // wave-uniform
  {
    int s = 6;
    if (slot6_is3) {
      int mt = tau6 >> 1, nt = tau6 & 1;
      int m = mt * 16 + r;
      int i = m / 20, j = m - 20 * i;
      iOff[s] = i * CHP; jOff[s] = j * CHP;
      mbase[s] = mt * 16; ntA[s] = nt;
    } else {
      int mt2 = (tau6 - 50) >> 1, nt = tau6 & 1;
      int m = mt2 * 16 + r;
      iOff[s] = m * CHP; jOff[s] = 0;   // AB rows 20..31 are zero => pad
      mbase[s] = mt2 * 16; ntA[s] = nt;
    }
  }

  v8f acc[7];
  const v8f vzero = {0.f,0.f,0.f,0.f,0.f,0.f,0.f,0.f};
  #pragma unroll
  for (int s = 0; s < 7; ++s) acc[s] = vzero;

  __syncthreads();

  for (int chunk = 0; chunk < NCHUNK; ++chunk) {
    // ---- stage chunk into LDS (closed forms from telescoped prefixes) ----
    int g = chunk * CH + tloc;
    if (g < LL - 1) {
      #pragma unroll
      for (int cc = 0; cc < 5; ++cc) {
        float x0 = Xb[g * CC + c0 + cc];
        float x1 = Xb[(g + 1) * CC + c0 + cc];
        float d  = x1 - x0;
        DB[(c0 + cc) * CHP + tloc] = d;
        AB[(c0 + cc) * CHP + tloc] = 0.5f * (x0 + x1) - xf[cc];
        BB[(c0 + cc) * CHP + tloc] = 0.5f * (x0 - xf[cc]) + d * (1.0f / 6.0f);
        RB[(c0 + cc) * CHP + tloc] = xl[cc] - x1;
      }
    } else {  // padded step: zero increment == identity step
      #pragma unroll
      for (int cc = 0; cc < 5; ++cc) {
        DB[(c0 + cc) * CHP + tloc] = 0.0f;
        AB[(c0 + cc) * CHP + tloc] = 0.0f;
        BB[(c0 + cc) * CHP + tloc] = 0.0f;
        RB[(c0 + cc) * CHP + tloc] = 0.0f;
      }
    }
    if (chunk + 1 < NCHUNK) {
      int gp = (chunk + 1) * CH + tloc;
      if (gp < LL) __builtin_prefetch(&Xb[gp * CC + c0], 0, 1);
    }
    __syncthreads();

    // ---- K loop: 16 k-steps of 4 timesteps each ----
    for (int ks = 0; ks < CH / 4; ++ks) {
      int tb = ks * 4 + 2 * half;     // t for frag VGPR0 (VGPR1 = tb+1)
      v2f Rf[2], Df[2];
      #pragma unroll
      for (int nt = 0; nt < 2; ++nt) {
        int n = nt * 16 + r;          // n>=20 reads zero padding
        Rf[nt][0] = RB[n * CHP + tb]; Rf[nt][1] = RB[n * CHP + tb + 1];
        Df[nt][0] = DB[n * CHP + tb]; Df[nt][1] = DB[n * CHP + tb + 1];
      }
      // slots 0..5: unconditional S3
      #pragma unroll
      for (int s = 0; s < 6; ++s) {
        v2f av; av[0] = AB[iOff[s] + tb]; av[1] = AB[iOff[s] + tb + 1];
        v2f dj; dj[0] = DB[jOff[s] + tb]; dj[1] = DB[jOff[s] + tb + 1];
        v2f bv; bv[0] = BB[iOff[s] + tb]; bv[1] = BB[iOff[s] + tb + 1];
        v2f u  = {av[0] * dj[0], av[1] * dj[1]};   // A_t[i]*d_t[j]
        v2f vv = {bv[0] * dj[0], bv[1] * dj[1]};   // B_t[i]*d_t[j]
        v2f br = ntA[s] ? Rf[1] : Rf[0];
        v2f bd = ntA[s] ? Df[1] : Df[0];
        acc[s] = __builtin_amdgcn_wmma_f32_16x16x4_f32(
            false, u,  false, br, (short)0, acc[s], false, false);
        acc[s] = __builtin_amdgcn_wmma_f32_16x16x4_f32(
            false, vv, false, bd, (short)0, acc[s], false, false);
      }
      // slot 6: wave-uniform S3/S2 choice
      {
        v2f av; av[0] = AB[iOff[6] + tb]; av[1] = AB[iOff[6] + tb + 1];
        v2f bd = ntA[6] ? Df[1] : Df[0];
        if (slot6_is3) {
          v2f dj; dj[0] = DB[jOff[6] + tb]; dj[1] = DB[jOff[6] + tb + 1];
          v2f bv; bv[0] = BB[iOff[6] + tb]; bv[1] = BB[iOff[6] + tb + 1];
          v2f u  = {av[0] * dj[0], av[1] * dj[1]};
          v2f vv = {bv[0] * dj[0], bv[1] * dj[1]};
          v2f br = ntA[6] ? Rf[1] : Rf[0];
          acc[6] = __builtin_amdgcn_wmma_f32_16x16x4_f32(
              false, u,  false, br, (short)0, acc[6], false, false);
          acc[6] = __builtin_amdgcn_wmma_f32_16x16x4_f32(
              false, vv, false, bd, (short)0, acc[6], false, false);
        } else {        // S2 tile: A_t[i] x d_t[n]
          acc[6] = __builtin_amdgcn_wmma_f32_16x16x4_f32(
              false, av, false, bd, (short)0, acc[6], false, false);
        }
      }
    }
    __syncthreads();
  }

  // ---- epilogue: signature into LDS (aliases dead chunk arrays) ----
  if (tloc == 0) {  // S1 = X[L-1] - X[0]
    #pragma unroll
    for (int cc = 0; cc < 5; ++cc) SIG[c0 + cc] = xl[cc] - xf[cc];
  }
  if (tid < 16) O10[tid] = 0.0f;
  #pragma unroll
  for (int s = 0; s < 6; ++s) {       // S3 tiles
    int n = ntA[s] * 16 + r;
    #pragma unroll
    for (int v = 0; v < 8; ++v) {
      int m = mbase[s] + v + 8 * half;   // C/D layout: M = v + 8*half
      if (n < 20) SIG[CC + CC * CC + m * 20 + n] = acc[s][v];
    }
  }
  {
    int n = ntA[6] * 16 + r;
    #pragma unroll
    for (int v = 0; v < 8; ++v) {
      int m = mbase[6] + v + 8 * half;
      float val = acc[6][v];
      if (slot6_is3) {
        if (n < 20) SIG[CC + CC * CC + m * 20 + n] = val;
      } else {
        if (m < 20 && n < 20) SIG[CC + m * 20 + n] = val;
      }
    }
  }
  __syncthreads();

  // ---- linear head: out[b, :] = sig . W^T + bias ----
  float part[NOUT];
  #pragma unroll
  for (int o = 0; o < NOUT; ++o) part[o] = 0.0f;
  for (int idx = tid; idx < SIGDIM; idx += 256) {
    float sv = SIG[idx];
    #pragma unroll
    for (int o = 0; o < NOUT; ++o) part[o] += W[o * SIGDIM + idx] * sv;
  }
  #pragma unroll
  for (int o = 0; o < NOUT; ++o) atomicAdd(&O10[o], part[o]);  // ds_add_f32
  __syncthreads();
  if (tid < NOUT) out[b * NOUT + tid] = O10[tid] + bias[tid];
}

extern "C" void kernel_launch(void* const* d_in, const int* in_sizes, int n_in,
                              void* d_out, int out_size, void* d_ws, size_t ws_size,
                              hipStream_t stream) {
  (void)n_in; (void)out_size; (void)d_ws; (void)ws_size;
  const float* X    = (const float*)d_in[0];
  const float* W    = (const float*)d_in[1];
  const float* bias = (const float*)d_in[2];
  float* out = (float*)d_out;
  int B = in_sizes[0] / (LL * CC);   // 32
  size_t shmem = (size_t)SMEM_FLOATS * sizeof(float);
  sig3_kernel<<<dim3(B), dim3(256), shmem, stream>>>(X, W, bias, out);
}